// DCNv4_41154376631108
// MI455X (gfx1250) — compile-verified
//
#include <hip/hip_runtime.h>
#include <hip/hip_bf16.h>
#include <math.h>

// ---------- problem constants (match reference) ----------
#define N_   2
#define C_   64
#define H_   192
#define W_   192
#define G_   4
#define DG_  16
#define P_   9
#define KPAD_ 112          // ceil(36*3/8)*8
#define HW_  (H_ * W_)
#define M_   (N_ * HW_)    // 73728 pixels

typedef __attribute__((ext_vector_type(2))) float v2f;
typedef __attribute__((ext_vector_type(8))) float v8f;

// =====================================================================
// Kernel 0: NCHW -> NHWC transpose of input (for channel-contiguous
// gathers in the deformable core).
// =====================================================================
__global__ void k_transpose_nhwc(const float* __restrict__ in, float* __restrict__ out) {
    int idx = blockIdx.x * blockDim.x + threadIdx.x;   // over N*HW*C, c innermost
    if (idx >= N_ * HW_ * C_) return;
    int c   = idx & (C_ - 1);
    int pix = idx >> 6;               // n*HW + hw
    int hw  = pix % HW_;
    int n   = pix / HW_;
    out[idx] = in[((size_t)(n * C_ + c)) * HW_ + hw];
}

// =====================================================================
// Kernel 1: depthwise 3x3 conv + bias on y, output written NHWC
// grid = N*C*H blocks of W threads (coalesced reads along w).
// =====================================================================
__global__ void k_dwconv(const float* __restrict__ y,
                         const float* __restrict__ wgt,
                         const float* __restrict__ bias,
                         float* __restrict__ ydw_nhwc) {
    int b = blockIdx.x;
    int h = b % H_;
    int c = (b / H_) & (C_ - 1);
    int n = b / (H_ * C_);
    int w = threadIdx.x;

    const float* yc = y + ((size_t)(n * C_ + c)) * HW_;
    float acc = bias[c];
#pragma unroll
    for (int dy = 0; dy < 3; ++dy) {
        int hh = h + dy - 1;
        if (hh < 0 || hh >= H_) continue;
#pragma unroll
        for (int dx = 0; dx < 3; ++dx) {
            int ww = w + dx - 1;
            if (ww < 0 || ww >= W_) continue;
            acc = fmaf(yc[hh * W_ + ww], wgt[c * 9 + dy * 3 + dx], acc);
        }
    }
    ydw_nhwc[(((size_t)(n * H_ + h) * W_ + w)) * C_ + c] = acc;
}

// =====================================================================
// Kernel 2: offset/mask projection GEMM via V_WMMA_F32_16X16X4_F32
//   offset_mask[M, 112] = ydw_nhwc[M, 64] @ om_weight[112,64]^T + om_bias
// Block: 128 threads = 4 waves; each wave owns one 16-row M tile and
// sweeps all 7 N-tiles. Weights+bias staged in LDS.
// f32 WMMA fragment layouts (ISA 7.12.2):
//   A 16x4 : lane L -> row M=L&15, kbase=2*(L>>4); regs hold K=kbase,kbase+1
//   B 4x16 : lane L -> col N=L&15, kbase=2*(L>>4); regs hold K=kbase,kbase+1
//   D 16x16: vgpr v, lane L -> M = v + 8*(L>>4), N = L&15
// =====================================================================
__global__ void k_om_gemm(const float* __restrict__ ydw,
                          const float* __restrict__ om_weight, // [112][64]
                          const float* __restrict__ om_bias,   // [112]
                          float* __restrict__ om_out) {        // [M][112]
    __shared__ float lds_w[C_ * KPAD_];   // lds_w[c][kout], 28 KB
    __shared__ float lds_b[KPAD_];

    int tid = threadIdx.x;
    // stage transposed weights: lds_w[c*112 + kout] = om_weight[kout*64 + c]
    for (int i = tid; i < C_ * KPAD_; i += 128) {
        int c    = i / KPAD_;
        int kout = i % KPAD_;
        lds_w[i] = om_weight[kout * C_ + c];
    }
    if (tid < KPAD_) lds_b[tid] = om_bias[tid];
    __syncthreads();

    const int wave = tid >> 5;
    const int lane = tid & 31;
    const int row  = lane & 15;
    const int half = lane >> 4;

    const int mtile = blockIdx.x * 4 + wave;   // 4608 M tiles total
    const int m0    = mtile * 16;
    const int m     = m0 + row;

    // preload all 16 A fragments (K = 0..63) for this wave's M tile
    v2f afrag[16];
    const v2f* ap = reinterpret_cast<const v2f*>(ydw + (size_t)m * C_ + 2 * half);
#pragma unroll
    for (int kk = 0; kk < 16; ++kk) afrag[kk] = ap[2 * kk];   // +4*kk floats

#pragma unroll
    for (int nt = 0; nt < 7; ++nt) {
        const int ncol = nt * 16 + row;
        float bv = lds_b[ncol];
        v8f acc;
#pragma unroll
        for (int v = 0; v < 8; ++v) acc[v] = bv;

#pragma unroll
        for (int kk = 0; kk < 16; ++kk) {
            const int kb = kk * 4 + 2 * half;
            v2f bfrag;
            bfrag[0] = lds_w[kb * KPAD_ + ncol];
            bfrag[1] = lds_w[(kb + 1) * KPAD_ + ncol];
            acc = __builtin_amdgcn_wmma_f32_16x16x4_f32(
                /*neg_a=*/false, afrag[kk], /*neg_b=*/false, bfrag,
                /*c_mod=*/(short)0, acc, /*reuse_a=*/false, /*reuse_b=*/false);
        }

#pragma unroll
        for (int v = 0; v < 8; ++v) {
            const int mm = m0 + v + 8 * half;
            om_out[(size_t)mm * KPAD_ + ncol] = acc[v];
        }
    }
}

// =====================================================================
// Kernel 3: deformable bilinear sampling core.
// One thread per (pixel, group, channel-quarter): computes 4 channels.
// =====================================================================
__global__ void k_dcn_core(const float* __restrict__ xt,   // NHWC input
                           const float* __restrict__ om,   // [M][112]
                           float* __restrict__ out) {      // NCHW output
    int t = blockIdx.x * blockDim.x + threadIdx.x;         // N*HW*G*4 threads
    int q   = t & 3;
    int g   = (t >> 2) & 3;
    int pix = t >> 4;                 // n*HW + h*W + w
    int w   = pix % W_;
    int h   = (pix / W_) % H_;
    int n   = pix / HW_;

    const float* omp = om + (size_t)pix * KPAD_ + g * 27;
    float offx[P_], offy[P_], msk[P_];
#pragma unroll
    for (int p = 0; p < P_; ++p) {
        offx[p] = omp[2 * p];
        offy[p] = omp[2 * p + 1];
        msk[p]  = omp[18 + p];
    }

    const float* xbase = xt + ((size_t)n * HW_) * C_ + g * DG_ + q * 4;

    float a0 = 0.f, a1 = 0.f, a2 = 0.f, a3 = 0.f;
#pragma unroll
    for (int p = 0; p < P_; ++p) {
        const float py = (float)(h - 1 + p / 3) + offy[p];
        const float px = (float)(w - 1 + p % 3) + offx[p];
        const float y0f = floorf(py);
        const float x0f = floorf(px);
        const float ly = py - y0f;
        const float lx = px - x0f;
        const int y0 = (int)y0f;
        const int x0 = (int)x0f;

        float s0 = 0.f, s1 = 0.f, s2 = 0.f, s3 = 0.f;
#pragma unroll
        for (int cy = 0; cy < 2; ++cy) {
#pragma unroll
            for (int cx = 0; cx < 2; ++cx) {
                const int yy = y0 + cy;
                const int xx = x0 + cx;
                const bool valid = (yy >= 0) & (yy < H_) & (xx >= 0) & (xx < W_);
                const int yc = min(max(yy, 0), H_ - 1);
                const int xc = min(max(xx, 0), W_ - 1);
                const float wy = cy ? ly : (1.f - ly);
                const float wx = cx ? lx : (1.f - lx);
                const float wz = valid ? (wy * wx) : 0.f;
                const float4 v =
                    *reinterpret_cast<const float4*>(xbase + (size_t)(yc * W_ + xc) * C_);
                s0 = fmaf(wz, v.x, s0);
                s1 = fmaf(wz, v.y, s1);
                s2 = fmaf(wz, v.z, s2);
                s3 = fmaf(wz, v.w, s3);
            }
        }
        a0 = fmaf(msk[p], s0, a0);
        a1 = fmaf(msk[p], s1, a1);
        a2 = fmaf(msk[p], s2, a2);
        a3 = fmaf(msk[p], s3, a3);
    }

    const int c0 = g * DG_ + q * 4;
    float* o = out + ((size_t)(n * C_ + c0)) * HW_ + h * W_ + w;
    o[0 * HW_] = a0;
    o[1 * HW_] = a1;
    o[2 * HW_] = a2;
    o[3 * HW_] = a3;
}

// =====================================================================
extern "C" void kernel_launch(void* const* d_in, const int* in_sizes, int n_in,
                              void* d_out, int out_size, void* d_ws, size_t ws_size,
                              hipStream_t stream) {
    const float* input     = (const float*)d_in[0];
    const float* y         = (const float*)d_in[1];
    const float* dw_weight = (const float*)d_in[2];
    const float* dw_bias   = (const float*)d_in[3];
    const float* om_weight = (const float*)d_in[4];
    const float* om_bias   = (const float*)d_in[5];
    float* out = (float*)d_out;

    // workspace layout
    float* xt  = (float*)d_ws;                  // NHWC input:  M*64 floats
    float* ydw = xt + (size_t)M_ * C_;          // NHWC dwconv: M*64 floats
    float* om  = ydw + (size_t)M_ * C_;         // offset/mask: M*112 floats

    // 0) NCHW -> NHWC transpose of input
    {
        int total = M_ * C_;
        k_transpose_nhwc<<<(total + 255) / 256, 256, 0, stream>>>(input, xt);
    }
    // 1) depthwise conv + bias (NHWC out)
    {
        k_dwconv<<<N_ * C_ * H_, W_, 0, stream>>>(y, dw_weight, dw_bias, ydw);
    }
    // 2) WMMA projection GEMM: om = ydw @ W^T + b
    {
        // 4608 M-tiles of 16 rows, 4 per block
        k_om_gemm<<<(M_ / 16) / 4, 128, 0, stream>>>(ydw, om_weight, om_bias, om);
    }
    // 3) deformable core
    {
        int total = M_ * G_ * 4;
        k_dcn_core<<<(total + 255) / 256, 256, 0, stream>>>(xt, om, out);
    }
}